// CombineSF_24747601559671
// MI455X (gfx1250) — compile-verified
//
#include <hip/hip_runtime.h>
#include <cstdint>
#include <cstddef>

// segment_sum scatter-add: out[tag[r], :] += data[r, :]
// D = 1024 f32 per row, 32768 rows, 16384 output segments.
// Memory-bound: ~256 MB of HBM traffic -> ~11 us floor at 23.3 TB/s.
// CDNA5 paths used:
//  - global_load_async_to_lds_b128 (ASYNCcnt) double-buffered row staging,
//    with th:TH_LOAD_NT so the one-shot 128MB input stream does not evict
//    the L2-resident 64MB output.
//  - global_atomic_add_f32 scatter resolved in L2 (output < 192MB L2).

#define COMBINE_D    1024
#define COMBINE_TPB  256     // 8 wave32 waves per block
#define COMBINE_RPB  8       // rows handled per block

__global__ __launch_bounds__(COMBINE_TPB)
void combine_zero_kernel(float4* __restrict__ out, int n4) {
    int i = blockIdx.x * COMBINE_TPB + threadIdx.x;
    const int stride = gridDim.x * COMBINE_TPB;
    float4 z; z.x = 0.f; z.y = 0.f; z.z = 0.f; z.w = 0.f;
    for (; i < n4; i += stride) out[i] = z;   // RT stores: seed output in L2
}

// One lane moves 16 bytes of a row into LDS via the async DMA path.
// SADDR form: global_load_async_to_lds_b128 vdst_lds, voffset, s[base:base+1]
// NT hint: input is streamed once; keep it out of the way of the output in L2.
__device__ __forceinline__
void async_row_slice_to_lds(unsigned lds_byte_off, const float* row_base,
                            unsigned byte_off) {
    unsigned long long sbase = (unsigned long long)(uintptr_t)row_base;
    asm volatile("global_load_async_to_lds_b128 %0, %1, %2 th:TH_LOAD_NT"
                 :: "v"(lds_byte_off), "v"(byte_off), "s"(sbase)
                 : "memory");
}

__global__ __launch_bounds__(COMBINE_TPB)
void combine_scatter_kernel(const float* __restrict__ data,
                            const int*   __restrict__ tags,
                            float*       __restrict__ out,
                            int nrows) {
    __shared__ float buf[2][COMBINE_D];

    const int tid  = threadIdx.x;
    const int row0 = blockIdx.x * COMBINE_RPB;
    if (row0 >= nrows) return;
    const int nr = (nrows - row0 < COMBINE_RPB) ? (nrows - row0) : COMBINE_RPB;

    const unsigned byte_off = (unsigned)tid * 16u;   // 16 B per lane
    // Flat shared pointers: low 32 bits are the LDS byte offset.
    unsigned lds_off[2];
    lds_off[0] = (unsigned)(uintptr_t)(&buf[0][tid * 4]);
    lds_off[1] = (unsigned)(uintptr_t)(&buf[1][tid * 4]);

    // Prime the pipeline with row 0.
    async_row_slice_to_lds(lds_off[0], data + (size_t)row0 * COMBINE_D, byte_off);

    for (int r = 0; r < nr; ++r) {
        if (r + 1 < nr) {
            // Prefetch next row into the other buffer, then wait for the
            // older transfer (async loads complete in order per wave).
            async_row_slice_to_lds(lds_off[(r + 1) & 1],
                                   data + (size_t)(row0 + r + 1) * COMBINE_D,
                                   byte_off);
            asm volatile("s_wait_asynccnt 1" ::: "memory");
        } else {
            asm volatile("s_wait_asynccnt 0" ::: "memory");
        }

        const int tag = tags[row0 + r];              // uniform broadcast load
        const float4 v = *(const float4*)(&buf[r & 1][tid * 4]);

        float* o = out + (size_t)tag * COMBINE_D + (size_t)(tid * 4);
        // Hardware global_atomic_add_f32; output is L2-resident (64MB < 192MB L2).
        unsafeAtomicAdd(o + 0, v.x);
        unsafeAtomicAdd(o + 1, v.y);
        unsafeAtomicAdd(o + 2, v.z);
        unsafeAtomicAdd(o + 3, v.w);
    }
}

extern "C" void kernel_launch(void* const* d_in, const int* in_sizes, int n_in,
                              void* d_out, int out_size, void* d_ws, size_t ws_size,
                              hipStream_t stream) {
    const float* data = (const float*)d_in[0];   // [P*N, D] f32
    const int*   tags = (const int*)d_in[1];     // [P*N] int32
    float*       out  = (float*)d_out;           // [load, D] f32

    const int nrows = in_sizes[1];               // P*N = 32768
    const int n4    = out_size / 4;              // float4 count

    combine_zero_kernel<<<2048, COMBINE_TPB, 0, stream>>>((float4*)out, n4);

    const int blocks = (nrows + COMBINE_RPB - 1) / COMBINE_RPB;
    combine_scatter_kernel<<<blocks, COMBINE_TPB, 0, stream>>>(data, tags, out, nrows);
}